// E2TensorProduct_FirstOrder_89979564851584
// MI455X (gfx1250) — compile-verified
//
#include <hip/hip_runtime.h>
#include <math.h>

// ---------------------------------------------------------------------------
// Problem constants (from reference): N1=N2=4096, K=16, H=8, C=128, M=9
// ---------------------------------------------------------------------------
#define CCH   128          // channels
#define MTOT  9            // (LMAX+1)^2
#define KNB   16           // neighbors
#define HH    8            // heads
#define ROWF  (MTOT*CCH)   // floats per node row-block (1152)
#define W3J_TOTAL 210      // 9+9+27+45+45+75

typedef __attribute__((ext_vector_type(2))) float v2f;
typedef __attribute__((ext_vector_type(8))) float v8f;

// ===========================================================================
// Kernel 0: device-side real Wigner-3j construction (e3nn convention),
// double precision, single thread (tiny). Writes 210 floats to workspace.
// ===========================================================================
struct cplx { double re, im; };
__device__ inline cplx cmul(cplx a, cplx b) { return {a.re*b.re - a.im*b.im, a.re*b.im + a.im*b.re}; }
__device__ inline cplx cconj(cplx a) { return {a.re, -a.im}; }

__device__ inline int imaxi(int a, int b) { return a > b ? a : b; }
__device__ inline int imini(int a, int b) { return a < b ? a : b; }
__device__ inline int iabsi(int a) { return a < 0 ? -a : a; }

__device__ double dfact(int n) { double r = 1.0; for (int i = 2; i <= n; ++i) r *= (double)i; return r; }

__device__ double su2_cg(int j1, int j2, int j3, int m1, int m2, int m3) {
  if (m3 != m1 + m2) return 0.0;
  if (j3 < iabsi(j1 - j2) || j3 > j1 + j2) return 0.0;
  double pref = sqrt((double)(2*j3+1) * dfact(j1+j2-j3) * dfact(j1-j2+j3) *
                     dfact(-j1+j2+j3) / dfact(j1+j2+j3+1));
  pref *= sqrt(dfact(j3+m3)*dfact(j3-m3)*dfact(j1-m1)*dfact(j1+m1)*dfact(j2-m2)*dfact(j2+m2));
  int vmin = imaxi(0, imaxi(j2 - j3 - m1, j1 - j3 + m2));
  int vmax = imini(j1 + j2 - j3, imini(j1 - m1, j2 + m2));
  double s = 0.0;
  for (int v = vmin; v <= vmax; ++v) {
    double d = dfact(v)*dfact(j1+j2-j3-v)*dfact(j1-m1-v)*dfact(j2+m2-v)*
               dfact(j3-j2+m1+v)*dfact(j3-j1-m2+v);
    s += ((v & 1) ? -1.0 : 1.0) / d;
  }
  return pref * s;
}

__device__ void build_q(int l, cplx q[5][5]) {
  for (int i = 0; i < 5; ++i) for (int j = 0; j < 5; ++j) q[i][j] = {0.0, 0.0};
  const double s2 = 0.70710678118654752440;
  for (int m = -l; m < 0; ++m) { q[l+m][l-m].re = s2; q[l+m][l+m].im = -s2; }
  q[l][l].re = 1.0;
  for (int m = 1; m <= l; ++m) {
    double sg = (m & 1) ? -1.0 : 1.0;
    q[l+m][l+m].re = sg * s2;
    q[l+m][l-m].im = sg * s2;
  }
  cplx ph = (l == 0) ? cplx{1.0, 0.0} : (l == 1 ? cplx{0.0, -1.0} : cplx{-1.0, 0.0});
  for (int i = 0; i < 5; ++i) for (int j = 0; j < 5; ++j) q[i][j] = cmul(ph, q[i][j]);
}

__device__ void compute_w3j_path(int l1, int l3, float* out) {
  const int n1 = 2*l1+1, n3 = 2*l3+1;
  double Cc[5][3][5];
  for (int i = 0; i < n1; ++i)
    for (int j = 0; j < 3; ++j)
      for (int k = 0; k < n3; ++k)
        Cc[i][j][k] = su2_cg(l1, 1, l3, i - l1, j - 1, k - l3);
  cplx Q1[5][5], Q2[5][5], Q3[5][5];
  build_q(l1, Q1); build_q(1, Q2); build_q(l3, Q3);
  double Cre[5][3][5], Cim[5][3][5];
  double sr = 0.0, si = 0.0;
  for (int a = 0; a < n1; ++a)
    for (int b = 0; b < 3; ++b)
      for (int c = 0; c < n3; ++c) {
        cplx acc = {0.0, 0.0};
        for (int i = 0; i < n1; ++i)
          for (int j = 0; j < 3; ++j)
            for (int k = 0; k < n3; ++k) {
              double w = Cc[i][j][k];
              if (w == 0.0) continue;
              cplx t = cmul(cmul(Q1[i][a], Q2[j][b]), cconj(Q3[k][c]));
              acc.re += t.re * w; acc.im += t.im * w;
            }
        Cre[a][b][c] = acc.re; Cim[a][b][c] = acc.im;
        sr += fabs(acc.re); si += fabs(acc.im);
      }
  bool useRe = (sr >= si);
  double norm2 = 0.0;
  for (int a = 0; a < n1; ++a)
    for (int b = 0; b < 3; ++b)
      for (int c = 0; c < n3; ++c) {
        double v = useRe ? Cre[a][b][c] : Cim[a][b][c];
        norm2 += v * v;
      }
  double scale = sqrt((double)(2*l3+1)) / sqrt(norm2);
  for (int a = 0; a < n1; ++a)
    for (int b = 0; b < 3; ++b)
      for (int c = 0; c < n3; ++c)
        out[(a*3 + b)*n3 + c] =
            (float)((useRe ? Cre[a][b][c] : Cim[a][b][c]) * scale);
}

__global__ void w3j_kernel(float* __restrict__ w3j) {
  if (threadIdx.x != 0 || blockIdx.x != 0) return;
  // PATHS = [(0,1),(1,0),(1,1),(1,2),(2,1),(2,2)]; offsets 0,9,18,45,90,135
  compute_w3j_path(0, 1, w3j + 0);
  compute_w3j_path(1, 0, w3j + 9);
  compute_w3j_path(1, 1, w3j + 18);
  compute_w3j_path(1, 2, w3j + 45);
  compute_w3j_path(2, 1, w3j + 90);
  compute_w3j_path(2, 2, w3j + 135);
}

// ===========================================================================
// so3_linear as WMMA f32 16x16x4 GEMM, templated on degree L so the row
// mapping uses compile-time division (L in {0,1,2}; DL = 2L+1 rows per node).
// Rows of the GEMM are (node, m) pairs of the L-group; weight is W[L]
// (128x128, row-major [c][d]). Block = 256 threads = 8 waves; each wave owns
// a 16-col tile, the block owns a 16-row tile; K=128 -> 32 WMMAs per wave.
// A frag (16x4 f32): lane<16 row=lane K={k,k+1}; lane>=16 row=lane-16 K={k+2,k+3}.
// B frag (4x16 f32): VGPR v, halfsel hs -> row K = k + v + 2*hs, col = lane&15.
// C/D (16x16 f32): VGPR v -> row = v + 8*hs, col = lane&15.
// IN_PLACE: Y aliases X; each 16-row tile is private to its block, and the
// barrier between the k-loop (reads) and the stores makes it race-free.
// ===========================================================================
template <int L, bool PATH_NORM, bool IN_PLACE>
__global__ __launch_bounds__(256)
void so3_linear_wmma(const float* __restrict__ X, const float* __restrict__ W,
                     const float* __restrict__ bias, float* __restrict__ Y) {
  constexpr int DL   = 2 * L + 1;
  constexpr int MOFF = L * L;
  constexpr float SCALE =
      (L == 0) ? 1.0f : (L == 1 ? 0.57735026918962576f    // 1/sqrt(3)
                                : 0.70710678118654752f);  // 1/sqrt(2)
  const int rt = blockIdx.x;             // 16-row tile within the L-group

  const int lane = threadIdx.x & 31;
  const int wv   = threadIdx.x >> 5;     // column tile 0..7
  const int lr   = lane & 15;
  const int hs   = lane >> 4;            // half-select

  // A row pointer for this lane (compile-time DL -> cheap div/mod)
  const int rA = rt * 16 + lr;
  const int nA = rA / DL, mA = rA % DL;
  const float* arow = X + (size_t)(nA * MTOT + MOFF + mA) * CCH + 2 * hs;
  const int col = wv * 16 + lr;
  const float* wbase = W + (size_t)L * CCH * CCH + col + (size_t)(2 * hs) * CCH;

  v8f acc = {};
  #pragma unroll 8
  for (int k = 0; k < CCH; k += 4) {
    v2f a, b;
    a.x = arow[k];
    a.y = arow[k + 1];
    const float* bp = wbase + (size_t)k * CCH;
    b.x = bp[0];
    b.y = bp[CCH];
    acc = __builtin_amdgcn_wmma_f32_16x16x4_f32(
        /*neg_a=*/false, a, /*neg_b=*/false, b,
        /*c_mod=*/(short)0, acc, /*reuse_a=*/false, /*reuse_b=*/false);
  }

  float bv = 0.0f;
  if constexpr (L == 0) bv = bias[col];  // l=0 rows are all m==0

  if constexpr (IN_PLACE) __syncthreads();  // all k-loop reads done before stores

  #pragma unroll
  for (int v = 0; v < 8; ++v) {
    const int r  = rt * 16 + v + 8 * hs;
    const int n  = r / DL, mm = r % DL;
    float val = acc[v];
    if constexpr (L == 0) val += bv;
    if constexpr (PATH_NORM) val *= SCALE;
    Y[(size_t)(n * MTOT + MOFF + mm) * CCH + col] = val;
  }
}

template <bool PATH_NORM, bool IN_PLACE>
static inline void launch_so3_linear(const float* X, const float* W, const float* bias,
                                     float* Y, int N, hipStream_t stream) {
  so3_linear_wmma<0, PATH_NORM, IN_PLACE><<<(N * 1) / 16, 256, 0, stream>>>(X, W, bias, Y);
  so3_linear_wmma<1, PATH_NORM, IN_PLACE><<<(N * 3) / 16, 256, 0, stream>>>(X, W, bias, Y);
  so3_linear_wmma<2, PATH_NORM, IN_PLACE><<<(N * 5) / 16, 256, 0, stream>>>(X, W, bias, Y);
}

// ===========================================================================
// Depthwise uvu tensor product helpers (per (node, channel) thread).
// ===========================================================================
template <int L1, int L3>
__device__ inline void tp_path(const float* __restrict__ C3, const float x[MTOT],
                               const float y[3], float wp, float out[MTOT]) {
  constexpr int N1 = 2*L1+1, N3 = 2*L3+1, O1 = L1*L1, O3 = L3*L3;
  #pragma unroll
  for (int a = 0; a < N1; ++a)
    #pragma unroll
    for (int b = 0; b < 3; ++b) {
      const float tv = x[O1 + a] * y[b] * wp;
      #pragma unroll
      for (int cc = 0; cc < N3; ++cc)
        out[O3 + cc] = fmaf(C3[(a*3 + b)*N3 + cc], tv, out[O3 + cc]);
    }
}

__device__ inline void tp_uvu_all(const float* __restrict__ s_w3j,
                                  const float x[MTOT], const float y[3],
                                  const float* __restrict__ tp_w, int c,
                                  float out[MTOT]) {
  tp_path<0,1>(s_w3j + 0,   x, y, tp_w[0*CCH + c], out);
  tp_path<1,0>(s_w3j + 9,   x, y, tp_w[1*CCH + c], out);
  tp_path<1,1>(s_w3j + 18,  x, y, tp_w[2*CCH + c], out);
  tp_path<1,2>(s_w3j + 45,  x, y, tp_w[3*CCH + c], out);
  tp_path<2,1>(s_w3j + 90,  x, y, tp_w[4*CCH + c], out);
  tp_path<2,2>(s_w3j + 135, x, y, tp_w[5*CCH + c], out);
}

// ===========================================================================
// Kernel 2: c2_src[n] = tp_uvu(exp_h1[n], yscale*exp_pos[n], tp_w)
// grid = N2 blocks x 128 threads (thread = channel)
// ===========================================================================
__global__ __launch_bounds__(128)
void tp_source_kernel(const float* __restrict__ exp_h1,
                      const float* __restrict__ exp_pos,
                      const float* __restrict__ tp_w,
                      const float* __restrict__ w3j,
                      float* __restrict__ c2_src, float yscale) {
  __shared__ float s_w3j[W3J_TOTAL];
  const int c = threadIdx.x;
  const int n = blockIdx.x;
  s_w3j[c] = w3j[c];
  if (c + 128 < W3J_TOTAL) s_w3j[c + 128] = w3j[c + 128];
  __syncthreads();

  float y[3] = {exp_pos[n*3 + 0] * yscale,
                exp_pos[n*3 + 1] * yscale,
                exp_pos[n*3 + 2] * yscale};
  float x[MTOT];
  const float* base = exp_h1 + (size_t)n * ROWF + c;
  #pragma unroll
  for (int m = 0; m < MTOT; ++m) x[m] = base[m * CCH];

  float out[MTOT] = {0.f,0.f,0.f,0.f,0.f,0.f,0.f,0.f,0.f};
  tp_uvu_all(s_w3j, x, y, tp_w, c, out);

  float* ob = c2_src + (size_t)n * ROWF + c;
  #pragma unroll
  for (int m = 0; m < MTOT; ++m) ob[m * CCH] = out[m];
}

// ===========================================================================
// Kernel 3a: per target node — gather+aggregate both streams, TP, diff.
//   agg1[m,c] = sum_j alpha[b,j,h(c)] * exp_h1[idx[b,j], m, c]
//   agg2[m,c] = sum_j alpha[b,j,h(c)] * c2_src[idx[b,j], m, c]
//   diff = tp_uvu(agg1, yscale*pos[b]) - agg2   -> written to d_out
// ===========================================================================
__global__ __launch_bounds__(128)
void gather_tp_diff_kernel(const float* __restrict__ exp_h1,
                           const float* __restrict__ c2_src,
                           const float* __restrict__ pos,
                           const float* __restrict__ alpha,
                           const int*   __restrict__ idx,
                           const float* __restrict__ tp_w,
                           const float* __restrict__ w3j,
                           float* __restrict__ diff_out, float yscale) {
  __shared__ float s_w3j[W3J_TOTAL];
  __shared__ float s_alpha[KNB * HH];   // [j][h]
  __shared__ int   s_idx[KNB];
  const int c = threadIdx.x;
  const int b = blockIdx.x;
  s_w3j[c] = w3j[c];
  if (c + 128 < W3J_TOTAL) s_w3j[c + 128] = w3j[c + 128];
  s_alpha[c] = alpha[(size_t)b * (KNB * HH) + c];
  if (c < KNB) s_idx[c] = idx[b * KNB + c];
  __syncthreads();

  const int h = c >> 4;                 // head = c / (C/H)
  float agg1[MTOT] = {0.f,0.f,0.f,0.f,0.f,0.f,0.f,0.f,0.f};
  float agg2[MTOT] = {0.f,0.f,0.f,0.f,0.f,0.f,0.f,0.f,0.f};
  #pragma unroll 4
  for (int j = 0; j < KNB; ++j) {
    const float w = s_alpha[j * HH + h];
    const size_t base = (size_t)s_idx[j] * ROWF + c;
    #pragma unroll
    for (int m = 0; m < MTOT; ++m) {
      agg1[m] = fmaf(w, exp_h1[base + m * CCH], agg1[m]);
      agg2[m] = fmaf(w, c2_src[base + m * CCH], agg2[m]);
    }
  }

  float y[3] = {pos[b*3 + 0] * yscale, pos[b*3 + 1] * yscale, pos[b*3 + 2] * yscale};
  float d1[MTOT] = {0.f,0.f,0.f,0.f,0.f,0.f,0.f,0.f,0.f};
  tp_uvu_all(s_w3j, agg1, y, tp_w, c, d1);

  float* ob = diff_out + (size_t)b * ROWF + c;
  #pragma unroll
  for (int m = 0; m < MTOT; ++m) ob[m * CCH] = d1[m] - agg2[m];
}

// ===========================================================================
// Launch
// ===========================================================================
extern "C" void kernel_launch(void* const* d_in, const int* in_sizes, int n_in,
                              void* d_out, int out_size, void* d_ws, size_t ws_size,
                              hipStream_t stream) {
  const float* pos     = (const float*)d_in[0];
  const float* exp_pos = (const float*)d_in[1];
  // d_in[2] = h (unused by the reference computation)
  const float* exp_h   = (const float*)d_in[3];
  const float* alpha   = (const float*)d_in[4];
  const int*   idx     = (const int*)  d_in[5];
  const float* w1_w    = (const float*)d_in[6];
  const float* w1_b    = (const float*)d_in[7];
  const float* w2_w    = (const float*)d_in[8];
  const float* w2_b    = (const float*)d_in[9];
  const float* tp_w    = (const float*)d_in[10];
  float* out = (float*)d_out;

  const int N1 = in_sizes[0] / 3;
  const int N2 = in_sizes[1] / 3;

  // workspace layout: [w3j 256f pad][exp_h1 N2*1152f][c2_src N2*1152f]
  float* w3j    = (float*)d_ws;
  float* exp_h1 = w3j + 256;
  float* c2_src = exp_h1 + (size_t)N2 * ROWF;

  // COEF1 * sqrt(3/(4*pi))
  const float yscale = (float)(2.04665350914 * 0.48860251190291992158);

  // k0: Wigner-3j tables (tiny, device-side, deterministic)
  w3j_kernel<<<1, 1, 0, stream>>>(w3j);

  // k1: exp_h1 = so3_linear(exp_h, w1)   [WMMA f32, 3 L-group GEMMs]
  launch_so3_linear<false, false>(exp_h, w1_w, w1_b, exp_h1, N2, stream);

  // k2: c2_src = tp_uvu(exp_h1, ey1, tp_w)
  tp_source_kernel<<<N2, 128, 0, stream>>>(exp_h1, exp_pos, tp_w, w3j, c2_src, yscale);

  // k3a: diff -> d_out
  gather_tp_diff_kernel<<<N1, 128, 0, stream>>>(
      exp_h1, c2_src, pos, alpha, idx, tp_w, w3j, out, yscale);

  // k3b: out = so3_linear(diff, w2) * PATH_NORM  [WMMA f32, in-place on d_out]
  launch_so3_linear<true, true>(out, w2_w, w2_b, out, N1, stream);
}